// NormalizedWeightsVarSizedElementReduce_50302656971013
// MI455X (gfx1250) — compile-verified
//
#include <hip/hip_runtime.h>
#include <math.h>

typedef __attribute__((ext_vector_type(2))) float v2f;
typedef __attribute__((ext_vector_type(4))) float v4f;
typedef __attribute__((ext_vector_type(8))) float v8f;

#define D 128

// ---------------------------------------------------------------------------
// Order-preserving float <-> uint encoding so atomicMax(u32) == float max.
//   f >= 0 : enc = bits | 0x80000000
//   f <  0 : enc = ~bits
// enc == 0 decodes to the minimum (acts as -inf identity for empty segments).
// ---------------------------------------------------------------------------
__device__ __forceinline__ unsigned enc_f32(float f) {
    unsigned b = __float_as_uint(f);
    return (b & 0x80000000u) ? ~b : (b | 0x80000000u);
}
__device__ __forceinline__ float dec_f32(unsigned e) {
    return __uint_as_float((e & 0x80000000u) ? (e ^ 0x80000000u) : ~e);
}

// ---------------------------------------------------------------------------
// Kernel 0: init workspace (smax encoded = 0 == -inf, ssum = 0, agg = 0)
// ---------------------------------------------------------------------------
__global__ void k_init(unsigned* __restrict__ smax, float* __restrict__ ssum,
                       float* __restrict__ agg, int S) {
    int i = blockIdx.x * blockDim.x + threadIdx.x;
    if (i < S) { smax[i] = 0u; ssum[i] = 0.0f; }
    if (i < S * D) agg[i] = 0.0f;
}

// ---------------------------------------------------------------------------
// Kernel 1: scores[i] = dot(E[i,:], w_att); segment max via encoded atomicMax.
// One wave per 32 consecutive elements; lane L holds w_att[4L..4L+3].
// E is streamed with non-temporal loads (512 MB, zero reuse) so the L2 keeps
// the hot intermediates (scores/agg/ssum/smax) instead.
// ---------------------------------------------------------------------------
__global__ void k_scores(const float* __restrict__ E, const int* __restrict__ seg,
                         const float* __restrict__ w_att, float* __restrict__ scores,
                         unsigned* __restrict__ smax, int N) {
    const int lane = threadIdx.x & 31;
    const int wave = blockIdx.x * (blockDim.x >> 5) + (threadIdx.x >> 5);
    const int EPW  = 32;
    const int base = wave * EPW;
    if (base >= N) return;

    const v4f wv = ((const v4f*)w_att)[lane];

    for (int t = 0; t < EPW; ++t) {
        int e = base + t;
        if (e >= N) break;
        v4f x = __builtin_nontemporal_load((const v4f*)(E + (size_t)e * D) + lane);
        float d = x.x * wv.x + x.y * wv.y + x.z * wv.z + x.w * wv.w;
        // wave32 butterfly reduction
        #pragma unroll
        for (int m = 16; m > 0; m >>= 1) d += __shfl_xor(d, m, 32);
        if (lane == 0) {
            scores[e] = d;
            atomicMax(&smax[seg[e]], enc_f32(d));
        }
    }
}

// ---------------------------------------------------------------------------
// Kernel 2: ssum[s] += exp(score - smax[s]); run-length combine over 8 sorted
// elements per thread before touching the atomic.
// ---------------------------------------------------------------------------
__global__ void k_expsum(const float* __restrict__ scores, const int* __restrict__ seg,
                         const unsigned* __restrict__ smax, float* __restrict__ ssum,
                         int N) {
    const int RUN = 8;
    int base = (blockIdx.x * blockDim.x + threadIdx.x) * RUN;
    if (base >= N) return;
    int cur = -1;
    float acc = 0.0f;
    for (int k = 0; k < RUN; ++k) {
        int e = base + k;
        if (e >= N) break;
        int s = seg[e];
        float v = __expf(scores[e] - dec_f32(smax[s]));
        if (s != cur) {
            if (cur >= 0) atomicAdd(&ssum[cur], acc);
            cur = s;
            acc = v;
        } else {
            acc += v;
        }
    }
    if (cur >= 0) atomicAdd(&ssum[cur], acc);
}

// ---------------------------------------------------------------------------
// Kernel 3: agg[s,:] += prob_i * E[i,:].  Wave per 128-element chunk; each lane
// owns 4 of the 128 dims in a float4 register accumulator, flushed with f32
// global atomics only at segment boundaries (segments are sorted).
// E again streamed non-temporally.
// ---------------------------------------------------------------------------
__global__ void k_agg(const float* __restrict__ E, const int* __restrict__ seg,
                      const float* __restrict__ scores, const unsigned* __restrict__ smax,
                      const float* __restrict__ ssum, float* __restrict__ agg, int N) {
    const int lane = threadIdx.x & 31;
    const int wave = blockIdx.x * (blockDim.x >> 5) + (threadIdx.x >> 5);
    const int CHUNK = 128;
    const int base = wave * CHUNK;
    if (base >= N) return;

    v4f acc = {0.f, 0.f, 0.f, 0.f};
    int cur = -1;

    for (int t = 0; t < CHUNK; ++t) {
        int e = base + t;
        if (e >= N) break;
        int s = seg[e];
        if (s != cur) {
            if (cur >= 0) {
                float* p = agg + (size_t)cur * D + lane * 4;
                atomicAdd(p + 0, acc.x);
                atomicAdd(p + 1, acc.y);
                atomicAdd(p + 2, acc.z);
                atomicAdd(p + 3, acc.w);
            }
            cur = s;
            acc = (v4f){0.f, 0.f, 0.f, 0.f};
        }
        float prob = __expf(scores[e] - dec_f32(smax[s])) / ssum[s];
        v4f x = __builtin_nontemporal_load((const v4f*)(E + (size_t)e * D) + lane);
        acc.x += prob * x.x;
        acc.y += prob * x.y;
        acc.z += prob * x.z;
        acc.w += prob * x.w;
    }
    if (cur >= 0) {
        float* p = agg + (size_t)cur * D + lane * 4;
        atomicAdd(p + 0, acc.x);
        atomicAdd(p + 1, acc.y);
        atomicAdd(p + 2, acc.z);
        atomicAdd(p + 3, acc.w);
    }
}

// ---------------------------------------------------------------------------
// Kernel 4: out[S,128] = agg[S,128] @ w_out^T via V_WMMA_F32_16X16X4_F32.
// One wave per 16-row strip; 8 column tiles x 32 K-steps (K=4 each).
// A layout (16x4 f32): lanes 0-15 -> K={0,1}, lanes 16-31 -> K={2,3}, M=lane&15.
// B layout (4x16 f32): mirrored; B[k][n] = w_out[n][k] -> contiguous float2
// loads from w_out rows (64 KB, WGP$-resident). C/D: VGPR j holds M=j
// (lanes 0-15) / M=j+8 (lanes 16-31). Output stored non-temporally (write-once).
// ---------------------------------------------------------------------------
__global__ void k_out_gemm(const float* __restrict__ agg, const float* __restrict__ wout,
                           float* __restrict__ out, int S) {
    const int lane = threadIdx.x & 31;
    const int wave = blockIdx.x * (blockDim.x >> 5) + (threadIdx.x >> 5);
    const int row0 = wave * 16;
    if (row0 >= S) return;

    const int half = lane >> 4;   // 0: K offsets {0,1}, 1: K offsets {2,3}
    const int l    = lane & 15;

    const float* arow = agg + (size_t)(row0 + l) * D;  // A: M = l

    for (int c = 0; c < 8; ++c) {                      // 8 column tiles of 16
        v8f acc = {};
        const float* brow = wout + (size_t)(c * 16 + l) * D;  // B: N = c*16 + l
        #pragma unroll
        for (int kb = 0; kb < 32; ++kb) {              // K = 128 in steps of 4
            v2f a = *(const v2f*)(arow + kb * 4 + half * 2);
            v2f b = *(const v2f*)(brow + kb * 4 + half * 2);
            acc = __builtin_amdgcn_wmma_f32_16x16x4_f32(
                /*neg_a=*/false, a, /*neg_b=*/false, b,
                /*c_mod=*/(short)0, acc, /*reuse_a=*/false, /*reuse_b=*/false);
        }
        // store D tile: VGPR j -> row (row0 + half*8 + j), col c*16 + l
        float* o = out + (size_t)(row0 + half * 8) * D + c * 16 + l;
        #pragma unroll
        for (int j = 0; j < 8; ++j)
            __builtin_nontemporal_store(acc[j], o + (size_t)j * D);
    }
}

// ---------------------------------------------------------------------------
// Host-side launcher
// ---------------------------------------------------------------------------
extern "C" void kernel_launch(void* const* d_in, const int* in_sizes, int n_in,
                              void* d_out, int out_size, void* d_ws, size_t ws_size,
                              hipStream_t stream) {
    const float* E     = (const float*)d_in[0];   // [N,128]
    const int*   seg   = (const int*)d_in[1];     // [N]
    const float* w_att = (const float*)d_in[3];   // [128]
    const float* w_out = (const float*)d_in[4];   // [128,128]
    float*       out   = (float*)d_out;           // [S,128]

    const int N = in_sizes[1];
    const int S = out_size / D;

    // workspace layout
    unsigned* smax   = (unsigned*)d_ws;                 // S u32
    float*    ssum   = (float*)(smax + S);              // S f32
    float*    agg    = ssum + S;                        // S*128 f32
    float*    scores = agg + (size_t)S * D;             // N f32

    // 0) init
    {
        int total = S * D;
        int blocks = (total + 255) / 256;
        k_init<<<blocks, 256, 0, stream>>>(smax, ssum, agg, S);
    }
    // 1) scores + segment max
    {
        int waves  = (N + 31) / 32;            // 32 elements per wave
        int blocks = (waves + 7) / 8;          // 8 waves (256 threads) per block
        k_scores<<<blocks, 256, 0, stream>>>(E, seg, w_att, scores, smax, N);
    }
    // 2) segment sum of exp
    {
        int threads = (N + 7) / 8;
        int blocks  = (threads + 255) / 256;
        k_expsum<<<blocks, 256, 0, stream>>>(scores, seg, smax, ssum, N);
    }
    // 3) attention-weighted segment aggregate
    {
        int waves  = (N + 127) / 128;
        int blocks = (waves + 7) / 8;
        k_agg<<<blocks, 256, 0, stream>>>(E, seg, scores, smax, ssum, agg, N);
    }
    // 4) out = agg @ w_out^T  (WMMA f32 16x16x4)
    {
        int waves  = (S + 15) / 16;
        int blocks = (waves + 7) / 8;
        k_out_gemm<<<blocks, 256, 0, stream>>>(agg, w_out, out, S);
    }
}